// ReconstructPatchImage_77300821394089
// MI455X (gfx1250) — compile-verified
//
#include <hip/hip_runtime.h>
#include <stdint.h>

// Problem constants (from the reference)
#define Bb   64
#define Cc   1024
#define HWp  196          // 14*14
#define Hd   14
#define CCT  32           // channel tile per block
#define NW   8            // waves per block (wave32)
#define TPB  256

// Dynamic-LDS layout (single dynamic allocation -> starts at LDS byte offset 0)
//   [0, 12288)               : async staging, 8 waves * 2 buffers * 6 inputs * 32 lanes * 4B
//   [12288, +196*33*4)       : accumulation tile, padded stride 33 (bank-conflict free)
#define STAGE_FLOATS   (NW * 2 * 6 * 32)               // 3072 floats = 12288 B
#define TILE_OFF       STAGE_FLOATS
#define TILE_STRIDE    33
#define LDS_FLOATS     (TILE_OFF + HWp * TILE_STRIDE)  // 3072 + 6468 = 9540
#define LDS_BYTES      (LDS_FLOATS * 4)                // 38160 B

static_assert(LDS_BYTES < 64 * 1024, "stay under default dynamic-LDS cap");

// inv5(q): position of row-major q in the top-left -> bottom-right diagonal scan.
__device__ __forceinline__ int inv5_of(int q) {
    int y = q / Hd, x = q % Hd;
    int s = y + x;
    int cum  = (s <= 13) ? (s * (s + 1)) / 2 : 196 - ((27 - s) * (28 - s)) / 2;
    int ymin = (s > 13) ? (s - 13) : 0;
    return cum + (y - ymin);
}

// Per-lane async global->LDS dword copy (CDNA5 GLOBAL_LOAD_ASYNC_TO_LDS_B32,
// GVS mode: reads the dword at the saddr base plus vaddr offset into the LDS address in vdst; tracked by ASYNCcnt).
__device__ __forceinline__ void async_g2l_b32(uint32_t lds_byte_off, uint32_t glb_byte_off,
                                              uint64_t base) {
    asm volatile("global_load_async_to_lds_b32 %0, %1, %2"
                 :: "v"(lds_byte_off), "v"(glb_byte_off), "s"(base)
                 : "memory");
}

// ASYNCcnt loads complete in order: waiting <=6 after issuing 12 guarantees
// the older group of 6 has landed in LDS.
__device__ __forceinline__ void wait_async6() {
    asm volatile("s_wait_asynccnt 0x6" ::: "memory");
}
__device__ __forceinline__ void wait_async0() {
    asm volatile("s_wait_asynccnt 0x0" ::: "memory");
}

__global__ __launch_bounds__(TPB) void ReconstructPatchImage_kernel(
    const float* __restrict__ in0,  // left_to_right
    const float* __restrict__ in1,  // right_to_left
    const float* __restrict__ in2,  // top_to_bottom
    const float* __restrict__ in3,  // bottom_to_top
    const float* __restrict__ in4,  // top_left_to_bottom_right
    const float* __restrict__ in5,  // bottom_right_to_top_left
    const float* __restrict__ in6,  // top_right_to_bottom_left
    const float* __restrict__ in7,  // bottom_left_to_top_right
    float* __restrict__ out)
{
    extern __shared__ float smem[];

    const int tid  = threadIdx.x;
    const int lane = tid & 31;
    const int wave = tid >> 5;

    const int blk = blockIdx.x;        // b * 32 + channel-tile index
    const int b   = blk >> 5;
    const int cc0 = (blk & 31) * CCT;

    const size_t batch = (size_t)b * (HWp * Cc);
    const uint64_t base0 = (uint64_t)(uintptr_t)(in0 + batch);
    const uint64_t base1 = (uint64_t)(uintptr_t)(in1 + batch);
    const uint64_t base4 = (uint64_t)(uintptr_t)(in4 + batch);
    const uint64_t base5 = (uint64_t)(uintptr_t)(in5 + batch);
    const uint64_t base6 = (uint64_t)(uintptr_t)(in6 + batch);
    const uint64_t base7 = (uint64_t)(uintptr_t)(in7 + batch);
    const float* __restrict__ p2 = in2 + batch;
    const float* __restrict__ p3 = in3 + batch;

    const uint32_t colB = (uint32_t)((cc0 + lane) * 4);

    // Staging slot byte base for (this wave, buffer buf): 6 segments of 128 B each.
    auto stage_base = [&](int buf) -> uint32_t {
        return (uint32_t)((((wave * 2 + buf) * 6) * 32 + lane) * 4);
    };

    // Issue the 6 row-permuted gathers for row q into staging buffer `buf`.
    auto issue_row = [&](int q, int buf) {
        const int y = q / Hd, x = q % Hd;
        const int j5 = inv5_of(q);
        const int P0 = q;
        const int P1 = 195 - q;
        const int P4 = j5;
        const int P5 = 195 - j5;
        const int P6 = inv5_of(y * Hd + (13 - x));   // inv5(mirror(q))
        const int P7 = inv5_of((13 - y) * Hd + x);   // inv5(mirror(195-q))
        const uint32_t sb = stage_base(buf);
        async_g2l_b32(sb + 0u * 128u, (uint32_t)(P0 * Cc * 4) + colB, base0);
        async_g2l_b32(sb + 1u * 128u, (uint32_t)(P1 * Cc * 4) + colB, base1);
        async_g2l_b32(sb + 2u * 128u, (uint32_t)(P4 * Cc * 4) + colB, base4);
        async_g2l_b32(sb + 3u * 128u, (uint32_t)(P5 * Cc * 4) + colB, base5);
        async_g2l_b32(sb + 4u * 128u, (uint32_t)(P6 * Cc * 4) + colB, base6);
        async_g2l_b32(sb + 5u * 128u, (uint32_t)(P7 * Cc * 4) + colB, base7);
    };

    // ---------------- Phase 1: pipelined gather + sum into LDS tile ----------------
    int buf = 0;
    issue_row(wave, buf);                      // prologue: first row's 6 loads in flight

    for (int q = wave; q < HWp; q += NW) {
        const int qn = q + NW;
        const bool more = (qn < HWp);
        if (more) issue_row(qn, buf ^ 1);      // keep 12 async loads in flight

        // col_major pair (flat transpose of a 14336x14 block per batch), computed
        // while the async loads are in flight.
        const int f  = q * Cc + cc0 + lane;    // output flat index within batch
        const int w2 = f / (Hd * Cc);          // f / 14336
        const int r  = f - w2 * (Hd * Cc);
        const int g  = r * Hd + w2;            // source flat index (ttb)
        float s = p2[g];
        const int pp = g >> 10, ch = g & 1023; // btt: flip HW axis before col_major
        s += p3[(195 - pp) * Cc + ch];

        if (more) wait_async6(); else wait_async0();

        const float* st = &smem[((wave * 2 + buf) * 6) * 32 + lane];
        s += st[0] + st[32] + st[64] + st[96] + st[128] + st[160];

        smem[TILE_OFF + q * TILE_STRIDE + lane] = s;
        buf ^= 1;
    }

    __syncthreads();

    // ---------------- Phase 2: coalesced transposed write-out ----------------
    // 32 channel rows; each wave writes 4 rows of 196 contiguous floats.
    for (int rr = 0; rr < 4; ++rr) {
        const int cof = wave * 4 + rr;
        float* __restrict__ op = out + ((size_t)b * Cc + (cc0 + cof)) * HWp;
        for (int j = lane; j < HWp; j += 32) {
            op[j] = smem[TILE_OFF + j * TILE_STRIDE + cof];
        }
    }
}

extern "C" void kernel_launch(void* const* d_in, const int* in_sizes, int n_in,
                              void* d_out, int out_size, void* d_ws, size_t ws_size,
                              hipStream_t stream) {
    (void)in_sizes; (void)n_in; (void)d_ws; (void)ws_size; (void)out_size;
    const dim3 grid(Bb * (Cc / CCT));   // 2048 blocks
    const dim3 block(TPB);
    ReconstructPatchImage_kernel<<<grid, block, LDS_BYTES, stream>>>(
        (const float*)d_in[0], (const float*)d_in[1],
        (const float*)d_in[2], (const float*)d_in[3],
        (const float*)d_in[4], (const float*)d_in[5],
        (const float*)d_in[6], (const float*)d_in[7],
        (float*)d_out);
}